// social_stgcn_67946382623599
// MI455X (gfx1250) — compile-verified
//
#include <hip/hip_runtime.h>

// GCLSTM forward for MI455X (gfx1250). fp32 throughout (matches reference);
// dense GEMMs use V_WMMA_F32_16X16X4_F32 (wave32, 16x16 tiles, K-step 4).

typedef __attribute__((ext_vector_type(2))) float v2f;
typedef __attribute__((ext_vector_type(8))) float v8f;

static __device__ __forceinline__ float sigm(float x) { return 1.0f / (1.0f + expf(-x)); }

// ---------------- degree + normalization ----------------
__global__ void k_degree(const int* __restrict__ dst, float* __restrict__ cnt, int E) {
  int t = blockIdx.x * blockDim.x + threadIdx.x;
  if (t < E) atomicAdd(&cnt[dst[t]], 1.0f);
}

__global__ void k_dinv(const float* __restrict__ cnt, float* __restrict__ dg,
                       float* __restrict__ dc, int N) {
  int i = blockIdx.x * blockDim.x + threadIdx.x;
  if (i >= N) return;
  float c = cnt[i];
  dg[i] = rsqrtf(c + 2.0f);                       // GCN improved: deg + 2 self-loop
  dc[i] = (c > 0.0f) ? rsqrtf(c) : 0.0f;          // Cheb: safe dinv, no self loops
}

// ------------- edge scatter: out[dst] += scale * dinv[src]*dinv[dst] * in[src] -------------
// thread = (edge, 4-feature group); 16 threads per edge cover D=64.
__global__ void k_scatter(const float* __restrict__ in, float* __restrict__ out,
                          const int* __restrict__ src, const int* __restrict__ dst,
                          const float* __restrict__ dinv, float scale, int E) {
  int t = blockIdx.x * blockDim.x + threadIdx.x;
  int e = t >> 4;
  if (e >= E) return;
  int f = (t & 15) << 2;
  int s = src[e], d = dst[e];
  float c = scale * dinv[s] * dinv[d];
  float4 v = *reinterpret_cast<const float4*>(in + (size_t)s * 64 + f);
  float* o = out + (size_t)d * 64 + f;
  atomicAdd(o + 0, c * v.x);
  atomicAdd(o + 1, c * v.y);
  atomicAdd(o + 2, c * v.z);
  atomicAdd(o + 3, c * v.w);
}

// ---------------- elementwise helpers ----------------
__global__ void k_negcopy(const float* __restrict__ in, float* __restrict__ out, int total) {
  int i = blockIdx.x * blockDim.x + threadIdx.x;
  if (i < total) out[i] = -in[i];
}

// h = relu(agg + 2*dinv_g^2 * xw + b)
__global__ void k_gcn_finalize(const float* __restrict__ agg, const float* __restrict__ xw,
                               const float* __restrict__ dg, const float* __restrict__ b,
                               float* __restrict__ h, int total) {
  int i = blockIdx.x * blockDim.x + threadIdx.x;
  if (i >= total) return;
  int node = i >> 6, col = i & 63;
  float d = dg[node];
  h[i] = fmaxf(agg[i] + 2.0f * d * d * xw[i] + b[col], 0.0f);
}

// LSTM combine with H=C=0 structure: Cn = sigmoid(G0)*tanh(G2); H = sigmoid(G3 + wc2*Cn)*tanh(Cn)
__global__ void k_combine(const float* __restrict__ G0, const float* __restrict__ G2,
                          const float* __restrict__ G3, const float* __restrict__ wc2,
                          float* __restrict__ H, int total) {
  int idx = blockIdx.x * blockDim.x + threadIdx.x;
  if (idx >= total) return;
  float ig = sigm(G0[idx]);
  float tg = tanhf(G2[idx]);
  float Cn = ig * tg;
  float og = sigm(G3[idx] + wc2[idx & 63] * Cn);
  H[idx] = og * tanhf(Cn);
}

// ---------------- WMMA GEMM: out[N x 64] = X[N x KIN] @ W[KIN x 64] ----------------
// 256 threads = 8 waves; each wave owns a 16-row x 64-col tile (4x 16x16 f32 accumulators).
template <int KIN>
__global__ __launch_bounds__(256) void k_gemm_n64(const float* __restrict__ X,
                                                  const float* __restrict__ W,
                                                  float* __restrict__ out, int mtiles) {
  __shared__ float w[KIN * 64];
  for (int i = threadIdx.x; i < (KIN * 64) / 4; i += 256)
    reinterpret_cast<float4*>(w)[i] = reinterpret_cast<const float4*>(W)[i];
  __syncthreads();

  int wave = threadIdx.x >> 5;
  int lane = threadIdx.x & 31;
  int tile = blockIdx.x * 8 + wave;
  if (tile >= mtiles) return;   // whole-wave exit: EXEC stays all-ones for WMMA
  int row0 = tile << 4;
  int half = lane >> 4;         // 0: K/M slots 0..1 ; 1: slots 2..3 (ISA 16x4 f32 layout)
  int l16  = lane & 15;

  v8f acc[4];
#pragma unroll
  for (int n = 0; n < 4; ++n) acc[n] = (v8f)0.0f;

  const float* xr = X + (size_t)(row0 + l16) * KIN + 2 * half;
#pragma unroll
  for (int k = 0; k < KIN; k += 4) {
    v2f a = *reinterpret_cast<const v2f*>(xr + k);
    int kb = k + 2 * half;
#pragma unroll
    for (int n = 0; n < 4; ++n) {
      int col = n * 16 + l16;
      v2f b = {w[kb * 64 + col], w[(kb + 1) * 64 + col]};
      acc[n] = __builtin_amdgcn_wmma_f32_16x16x4_f32(false, a, false, b, (short)0,
                                                     acc[n], false, false);
    }
  }

  float* orow = out + (size_t)(row0 + half * 8) * 64;
#pragma unroll
  for (int n = 0; n < 4; ++n)
#pragma unroll
    for (int v = 0; v < 8; ++v)
      orow[(size_t)v * 64 + n * 16 + l16] = acc[n][v];
}

// ------- fused Cheb gate: out = X@W0 + Z1@W1 + Z2@W2 + (bx+bh+bg)  (gh(H=0) == bh) -------
__global__ __launch_bounds__(256) void k_cheb_gate(const float* __restrict__ X,
                                                   const float* __restrict__ Z1,
                                                   const float* __restrict__ Z2,
                                                   const float* __restrict__ W3,  // 3 x 64 x 64
                                                   const float* __restrict__ bx,
                                                   const float* __restrict__ bh,
                                                   const float* __restrict__ bg,
                                                   float* __restrict__ out, int mtiles) {
  __shared__ float w[3 * 64 * 64];   // 48 KB (WGP has 320 KB LDS)
  for (int i = threadIdx.x; i < (3 * 64 * 64) / 4; i += 256)
    reinterpret_cast<float4*>(w)[i] = reinterpret_cast<const float4*>(W3)[i];
  __syncthreads();

  int wave = threadIdx.x >> 5;
  int lane = threadIdx.x & 31;
  int tile = blockIdx.x * 8 + wave;
  if (tile >= mtiles) return;
  int row0 = tile << 4;
  int half = lane >> 4;
  int l16  = lane & 15;

  v8f acc[4];
#pragma unroll
  for (int n = 0; n < 4; ++n) {
    int col = n * 16 + l16;
    acc[n] = (v8f)(bx[col] + bh[col] + bg[col]);   // fold all biases into C
  }

  const float* xr  = X  + (size_t)(row0 + l16) * 64 + 2 * half;
  const float* z1r = Z1 + (size_t)(row0 + l16) * 64 + 2 * half;
  const float* z2r = Z2 + (size_t)(row0 + l16) * 64 + 2 * half;

#pragma unroll 4
  for (int k = 0; k < 64; k += 4) {
    v2f a0 = *reinterpret_cast<const v2f*>(xr + k);
    v2f a1 = *reinterpret_cast<const v2f*>(z1r + k);
    v2f a2 = *reinterpret_cast<const v2f*>(z2r + k);
    int kb = k + 2 * half;
#pragma unroll
    for (int n = 0; n < 4; ++n) {
      int col = n * 16 + l16;
      v2f b0 = {w[kb * 64 + col],        w[(kb + 1) * 64 + col]};
      v2f b1 = {w[4096 + kb * 64 + col], w[4096 + (kb + 1) * 64 + col]};
      v2f b2 = {w[8192 + kb * 64 + col], w[8192 + (kb + 1) * 64 + col]};
      acc[n] = __builtin_amdgcn_wmma_f32_16x16x4_f32(false, a0, false, b0, (short)0, acc[n], false, false);
      acc[n] = __builtin_amdgcn_wmma_f32_16x16x4_f32(false, a1, false, b1, (short)0, acc[n], false, false);
      acc[n] = __builtin_amdgcn_wmma_f32_16x16x4_f32(false, a2, false, b2, (short)0, acc[n], false, false);
    }
  }

  float* orow = out + (size_t)(row0 + half * 8) * 64;
#pragma unroll
  for (int n = 0; n < 4; ++n)
#pragma unroll
    for (int v = 0; v < 8; ++v)
      orow[(size_t)v * 64 + n * 16 + l16] = acc[n][v];
}

// ---------------- final: out = relu(H) @ lin_w (64x3) + lin_b ----------------
__global__ __launch_bounds__(256) void k_final_linear(const float* __restrict__ H,
                                                      const float* __restrict__ lw,
                                                      const float* __restrict__ lb,
                                                      float* __restrict__ out, int N) {
  __shared__ float w[192];
  __shared__ float b[3];
  if (threadIdx.x < 192) w[threadIdx.x] = lw[threadIdx.x];
  if (threadIdx.x < 3) b[threadIdx.x] = lb[threadIdx.x];
  __syncthreads();
  int node = blockIdx.x * blockDim.x + threadIdx.x;
  if (node >= N) return;
  const float* hr = H + (size_t)node * 64;
  float a0 = b[0], a1 = b[1], a2 = b[2];
#pragma unroll
  for (int d = 0; d < 64; ++d) {
    float hv = fmaxf(hr[d], 0.0f);
    a0 = fmaf(hv, w[d * 3 + 0], a0);
    a1 = fmaf(hv, w[d * 3 + 1], a1);
    a2 = fmaf(hv, w[d * 3 + 2], a2);
  }
  out[node * 3 + 0] = a0;
  out[node * 3 + 1] = a1;
  out[node * 3 + 2] = a2;
}

static inline int cdiv(int a, int b) { return (a + b - 1) / b; }

extern "C" void kernel_launch(void* const* d_in, const int* in_sizes, int n_in,
                              void* d_out, int out_size, void* d_ws, size_t ws_size,
                              hipStream_t stream) {
  (void)n_in; (void)out_size; (void)ws_size;
  const int N = in_sizes[0] / 16;
  const int E = in_sizes[1] / 2;
  const int mtiles = cdiv(N, 16);

  const float* x      = (const float*)d_in[0];
  const int*   ei     = (const int*)d_in[1];
  const int*   src    = ei;
  const int*   dst    = ei + E;
  const float* w_gcn1 = (const float*)d_in[2];
  const float* b_gcn1 = (const float*)d_in[3];
  const float* w_gcn2 = (const float*)d_in[4];
  const float* b_gcn2 = (const float*)d_in[5];
  const float* lin_w  = (const float*)d_in[24];
  const float* lin_b  = (const float*)d_in[25];
  float* out = (float*)d_out;

  // workspace carve (all fp32)
  float* p = (float*)d_ws;
  float* cnt   = p; p += N;
  float* dinvg = p; p += N;
  float* dinvc = p; p += N;
  const size_t NF = (size_t)N * 64;
  float* A  = p; p += NF;   // xw / gemm out
  float* B  = p; p += NF;   // scatter accumulator
  float* H  = p; p += NF;   // current activation
  float* Z1 = p; p += NF;
  float* Z2 = p; p += NF;
  float* G0 = p; p += NF;
  float* G2 = p; p += NF;
  float* G3 = p; p += NF;

  const int T = 256;
  const int gemm_blocks = cdiv(mtiles, 8);
  const int sc_blocks   = cdiv(E * 16, T);
  const int el_blocks   = cdiv((int)NF, T);

  // degree + normalizations
  hipMemsetAsync(cnt, 0, (size_t)N * 4, stream);
  k_degree<<<cdiv(E, T), T, 0, stream>>>(dst, cnt, E);
  k_dinv<<<cdiv(N, T), T, 0, stream>>>(cnt, dinvg, dinvc, N);

  // GCN1: xw = x@W1 ; agg ; h = relu(agg + 2*dinv^2*xw + b)
  k_gemm_n64<16><<<gemm_blocks, T, 0, stream>>>(x, w_gcn1, A, mtiles);
  hipMemsetAsync(B, 0, NF * 4, stream);
  k_scatter<<<sc_blocks, T, 0, stream>>>(A, B, src, dst, dinvg, 1.0f, E);
  k_gcn_finalize<<<el_blocks, T, 0, stream>>>(B, A, dinvg, b_gcn1, H, (int)NF);

  // GCN2
  k_gemm_n64<64><<<gemm_blocks, T, 0, stream>>>(H, w_gcn2, A, mtiles);
  hipMemsetAsync(B, 0, NF * 4, stream);
  k_scatter<<<sc_blocks, T, 0, stream>>>(A, B, src, dst, dinvg, 1.0f, E);
  k_gcn_finalize<<<el_blocks, T, 0, stream>>>(B, A, dinvg, b_gcn2, H, (int)NF);

  // 3 GCLSTM layers (H=C=0 inside each cell; f-gate unused since C=0)
  for (int l = 0; l < 3; ++l) {
    const float* wx = (const float*)d_in[6 + 6 * l];   // (4,3,64,64)
    const float* bx = (const float*)d_in[7 + 6 * l];   // (4,64)
    const float* bh = (const float*)d_in[9 + 6 * l];   // (4,64)  == gh(H=0)
    const float* wc = (const float*)d_in[10 + 6 * l];  // (3,64)
    const float* bg = (const float*)d_in[11 + 6 * l];  // (4,64)

    // Z1 = lhat(X) = -Anorm X ;  Z2 = 2*lhat(Z1) - X = -2*Anorm Z1 - X
    hipMemsetAsync(Z1, 0, NF * 4, stream);
    k_scatter<<<sc_blocks, T, 0, stream>>>(H, Z1, src, dst, dinvc, -1.0f, E);
    k_negcopy<<<el_blocks, T, 0, stream>>>(H, Z2, (int)NF);
    k_scatter<<<sc_blocks, T, 0, stream>>>(Z1, Z2, src, dst, dinvc, -2.0f, E);

    // gates i (g=0), c~ (g=2), o (g=3); gate stride in wx is 3*64*64 = 12288
    k_cheb_gate<<<gemm_blocks, T, 0, stream>>>(H, Z1, Z2, wx + 0 * 12288,
                                               bx + 0,   bh + 0,   bg + 0,   G0, mtiles);
    k_cheb_gate<<<gemm_blocks, T, 0, stream>>>(H, Z1, Z2, wx + 2 * 12288,
                                               bx + 128, bh + 128, bg + 128, G2, mtiles);
    k_cheb_gate<<<gemm_blocks, T, 0, stream>>>(H, Z1, Z2, wx + 3 * 12288,
                                               bx + 192, bh + 192, bg + 192, G3, mtiles);

    k_combine<<<el_blocks, T, 0, stream>>>(G0, G2, G3, wc + 128, H, (int)NF);
  }

  k_final_linear<<<cdiv(N, T), T, 0, stream>>>(H, lin_w, lin_b, out, N);
}